// SubmanifoldAttention_42202348650939
// MI455X (gfx1250) — compile-verified
//
#include <hip/hip_runtime.h>
#include <hip/hip_bf16.h>

// SubmanifoldAttention for MI455X (gfx1250, wave32).
//
// Only prob row j=127 survives the reference, so the work is: logm of all
// K/V bimaps (8192 SPD 48x48 matrices) + 32 Q bimaps, a 128-wide softmax mix
// per batch, and 32 expm's. logm/expm are recast as matmul-only iterations
// (coupled Newton-Schulz sqrt + Taylor series) so the FLOPs run on
// V_WMMA_F32_16X16X4_F32 (fp32 needed for log/exp spectra).
//
// Data movement: each block's unique X (96x96 f32, 36 KB) is staged into LDS
// once via GLOBAL_LOAD_ASYNC_TO_LDS_B128 (ASYNCcnt path), then reused by all
// WMMA B-operand reads. W matrices are tiny, shared by all blocks, L2-resident.
//
// Workspace (float): logK[4096][2304] | logV[4096][2304] | logQ[32][2304]
// = 75.8 MB (fits in the 192 MB L2).

typedef __attribute__((ext_vector_type(2))) float v2f;
typedef __attribute__((ext_vector_type(8))) float v8f;

#define DIN 96
#define DD  48
#define DD2 2304         // 48*48
#define M2  128          // concatenated sequence length
#define NS_LEVELS 4      // number of matrix square roots (2^4 scaling)
#define NS_ITERS  7      // Newton-Schulz iterations per sqrt
#define LOG_TERMS 10     // Taylor degree for log(I+E)
#define EXP_DEG   8      // Taylor degree for exp
#define EXP_SQ    5      // squarings for expm

// ---------------- WMMA helpers (f32 16x16x4) ----------------
// A tile 16x4 (MxK): lane L holds m=L%16, k0=2*(L/16): {A[m][k0], A[m][k0+1]}
// B tile 4x16 (KxN): lane L holds n=L%16, k0=2*(L/16): {B[k0][n], B[k0+1][n]}
// C/D 16x16: vgpr r, lanes 0-15 -> (M=r, N=lane); lanes 16-31 -> (M=r+8).

static __device__ inline v8f wmma4(v2f a, v2f b, v8f c) {
  return __builtin_amdgcn_wmma_f32_16x16x4_f32(
      /*neg_a=*/false, a, /*neg_b=*/false, b,
      /*c_mod=*/(short)0, c, /*reuse_a=*/false, /*reuse_b=*/false);
}

static __device__ inline v2f ld_a(const float* P, int ld, int r0, int c0, int lane) {
  int m = lane & 15, k0 = (lane >> 4) << 1;
  const float* p = P + (size_t)(r0 + m) * ld + c0 + k0;
  v2f a; a.x = p[0]; a.y = p[1];
  return a;
}

static __device__ inline v2f ld_b(const float* P, int ld, int r0, int c0, int lane) {
  int n = lane & 15, k0 = (lane >> 4) << 1;
  v2f b;
  b.x = P[(size_t)(r0 + k0) * ld + c0 + n];
  b.y = P[(size_t)(r0 + k0 + 1) * ld + c0 + n];
  return b;
}

// tile of P^T at (r0,c0): element (k,n) = P[c0+n][r0+k]
static __device__ inline v2f ld_bT(const float* P, int ld, int r0, int c0, int lane) {
  int n = lane & 15, k0 = (lane >> 4) << 1;
  const float* p = P + (size_t)(c0 + n) * ld + r0 + k0;
  v2f b; b.x = p[0]; b.y = p[1];
  return b;
}

static __device__ inline void st_c(float* Dst, int ld, int r0, int c0, int lane, v8f c) {
  int n = lane & 15, mh = (lane >> 4) << 3;
#pragma unroll
  for (int r = 0; r < 8; ++r)
    Dst[(size_t)(r0 + mh + r) * ld + c0 + n] = c[r];
}

// D = A*B, all 48x48 (ld=48), one wave, LDS buffers.
static __device__ inline void mm48(float* Dm, const float* Am, const float* Bm, int lane) {
  __syncthreads();
#pragma unroll
  for (int ti = 0; ti < 3; ++ti) {
#pragma unroll
    for (int tj = 0; tj < 3; ++tj) {
      v8f c = {};
#pragma unroll
      for (int k = 0; k < DD; k += 4) {
        v2f a = ld_a(Am, DD, ti * 16, k, lane);
        v2f b = ld_b(Bm, DD, k, tj * 16, lane);
        c = wmma4(a, b, c);
      }
      st_c(Dm, DD, ti * 16, tj * 16, lane, c);
    }
  }
  __syncthreads();
}

// ---------------- wave32 reductions ----------------
static __device__ inline float wave_sum(float v) {
#pragma unroll
  for (int m = 16; m > 0; m >>= 1) v += __shfl_xor(v, m, 32);
  return v;
}
static __device__ inline float wave_max(float v) {
#pragma unroll
  for (int m = 16; m > 0; m >>= 1) v = fmaxf(v, __shfl_xor(v, m, 32));
  return v;
}

// Async-stage `bytes` (multiple of 512) from global `src` into LDS at `dst`
// using GLOBAL_LOAD_ASYNC_TO_LDS_B128 (GVS mode: saddr base + 32-bit voffset).
// Generic LDS pointer low 32 bits == workgroup LDS offset (ISA 10.2 aperture).
static __device__ inline void async_stage(float* dst, const float* src,
                                          int bytes, int lane) {
  uint32_t lds0 = (uint32_t)(uintptr_t)dst;
  uint64_t g0 = (uint64_t)(uintptr_t)src;
  for (int it = 0; it < bytes; it += 512) {
    uint32_t off = (uint32_t)(lane << 4) + (uint32_t)it;
    uint32_t ldsa = lds0 + off;
    asm volatile("global_load_async_to_lds_b128 %0, %1, %2"
                 :: "v"(ldsa), "v"(off), "s"(g0)
                 : "memory");
  }
  asm volatile("s_wait_asynccnt 0x0" ::: "memory");
}

// =====================================================================
// Kernel 1: per-matrix  S = W X W^T  followed by  logm(S).
// Blocks 0..4095 -> K(b,i); 4096..8191 -> V(b,i); 8192..8223 -> Q(b, j=127).
// One wave32 per block.
// =====================================================================
__global__ void __launch_bounds__(32)
bimap_logm_kernel(const float* __restrict__ x1, const float* __restrict__ x2,
                  const float* __restrict__ wk1, const float* __restrict__ wk2,
                  const float* __restrict__ wv1, const float* __restrict__ wv2,
                  const float* __restrict__ wq2,
                  float* __restrict__ logK, float* __restrict__ logV,
                  float* __restrict__ logQ) {
  // LDS: A(2304) | region R(13824):
  //   stage1: T(48x96=4608) + Xs(96x96=9216)
  //   stage2: Y,Z,YN,ZN,E (5 x 2304 = 11520)   (overlays stage1)
  __shared__ float sh[DD2 + 13824];
  float* Abuf = sh;
  float* Rr = sh + DD2;

  const int lane = threadIdx.x;
  const int id = blockIdx.x;

  const float* X;
  const float* W;
  float* outp;
  if (id < 4096) {
    int b = id >> 7, i = id & 127;
    X = (i < 64 ? x1 : x2) + ((size_t)(b * 64 + (i & 63))) * (DIN * DIN);
    W = (i < 64 ? wk1 : wk2);
    outp = logK + (size_t)id * DD2;
  } else if (id < 8192) {
    int t = id - 4096;
    int b = t >> 7, i = t & 127;
    X = (i < 64 ? x1 : x2) + ((size_t)(b * 64 + (i & 63))) * (DIN * DIN);
    W = (i < 64 ? wv1 : wv2);
    outp = logV + (size_t)t * DD2;
  } else {
    int b = id - 8192;
    X = x2 + ((size_t)(b * 64 + 63)) * (DIN * DIN);
    W = wq2;
    outp = logQ + (size_t)b * DD2;
  }

  // warm W in caches (global_prefetch_b8); shared by thousands of blocks
  __builtin_prefetch(W, 0, 1);

  // ---- async-stage this block's X (36 KB) into LDS once ----
  float* T = Rr;          // 4608 floats
  float* Xs = Rr + 4608;  // 9216 floats
  async_stage(Xs, X, DIN * DIN * 4, lane);
  __syncthreads();

  // ---- T = W @ X  (48x96): A from global W, B from LDS Xs ----
#pragma unroll
  for (int ti = 0; ti < 3; ++ti) {
#pragma unroll
    for (int tj = 0; tj < 6; ++tj) {
      v8f c = {};
#pragma unroll 4
      for (int k = 0; k < DIN; k += 4) {
        v2f a = ld_a(W, DIN, ti * 16, k, lane);
        v2f b = ld_b(Xs, DIN, k, tj * 16, lane);
        c = wmma4(a, b, c);
      }
      st_c(T, DIN, ti * 16, tj * 16, lane, c);
    }
  }
  __syncthreads();

  // ---- S = T @ W^T  (48x48) -> Abuf ----
#pragma unroll
  for (int ti = 0; ti < 3; ++ti) {
#pragma unroll
    for (int tj = 0; tj < 3; ++tj) {
      v8f c = {};
#pragma unroll 4
      for (int k = 0; k < DIN; k += 4) {
        v2f a = ld_a(T, DIN, ti * 16, k, lane);
        v2f b = ld_bT(W, DIN, k, tj * 16, lane);
        c = wmma4(a, b, c);
      }
      st_c(Abuf, DD, ti * 16, tj * 16, lane, c);
    }
  }
  __syncthreads();

  // ---- normalize by trace: A <- S/c, logm(S) = logm(A) + log(c) I ----
  float tr = 0.0f;
  for (int d = lane; d < DD; d += 32) tr += Abuf[d * (DD + 1)];
  tr = wave_sum(tr);
  float cnorm = fmaxf(tr, 1e-30f);
  float inv = 1.0f / cnorm;
  float logc = logf(cnorm);
  for (int t = lane; t < DD2; t += 32) Abuf[t] *= inv;
  __syncthreads();

  // ---- NS_LEVELS coupled Newton-Schulz square roots (matmul only) ----
  for (int l = 0; l < NS_LEVELS; ++l) {
    float* Y = Rr;
    float* Z = Rr + DD2;
    float* YN = Rr + 2 * DD2;
    float* ZN = Rr + 3 * DD2;
    float* E = Rr + 4 * DD2;
    for (int t = lane; t < DD2; t += 32) {
      Y[t] = Abuf[t];
      Z[t] = (t % (DD + 1) == 0) ? 1.0f : 0.0f;
    }
    __syncthreads();
    for (int it = 0; it < NS_ITERS; ++it) {
      mm48(E, Z, Y, lane); // E = Z@Y
      for (int t = lane; t < DD2; t += 32)
        E[t] = ((t % (DD + 1) == 0) ? 1.5f : 0.0f) - 0.5f * E[t];
      __syncthreads();
      mm48(YN, Y, E, lane); // Y' = Y E
      mm48(ZN, E, Z, lane); // Z' = E Z
      float* tp;
      tp = Y; Y = YN; YN = tp;
      tp = Z; Z = ZN; ZN = tp;
    }
    for (int t = lane; t < DD2; t += 32) Abuf[t] = Y[t];
    __syncthreads();
  }

  // ---- log(A) via Taylor on E = A - I (Horner, one matmul per term) ----
  {
    float* Em = Rr;
    float* G = Rr + DD2;
    float* GN = Rr + 2 * DD2;
    const float cN = ((LOG_TERMS & 1) ? 1.0f : -1.0f) / (float)LOG_TERMS;
    for (int t = lane; t < DD2; t += 32) {
      float iv = (t % (DD + 1) == 0) ? 1.0f : 0.0f;
      Em[t] = Abuf[t] - iv;
      G[t] = iv * cN;
    }
    __syncthreads();
    for (int n = LOG_TERMS - 1; n >= 1; --n) {
      mm48(GN, Em, G, lane);
      float cn = ((n & 1) ? 1.0f : -1.0f) / (float)n;
      for (int t = lane; t < DD2; t += 32)
        GN[t] += (t % (DD + 1) == 0) ? cn : 0.0f;
      __syncthreads();
      float* tp = G; G = GN; GN = tp;
    }
    mm48(GN, Em, G, lane); // GN = log(A_normalized)

    const float scale = (float)(1 << NS_LEVELS);
    for (int t = lane; t < DD2; t += 32)
      outp[t] = scale * GN[t] + ((t % (DD + 1) == 0) ? logc : 0.0f);
  }
}

// =====================================================================
// Kernel 2: per batch b — scores over i, softmax, mix logV, expm, store.
// One wave32 per block, 32 blocks.
// =====================================================================
__global__ void __launch_bounds__(32)
attn_expm_kernel(const float* __restrict__ logK, const float* __restrict__ logV,
                 const float* __restrict__ logQ, float* __restrict__ out) {
  __shared__ float sh2[3 * DD2 + M2];
  float* Xb = sh2;             // scaled mixed matrix
  float* G = sh2 + DD2;
  float* GN = sh2 + 2 * DD2;
  float* scores = sh2 + 3 * DD2;

  const int lane = threadIdx.x;
  const int b = blockIdx.x;

  // stage logQ[b] in LDS (reuse G)
  float* qb = G;
  for (int t = lane; t < DD2; t += 32) qb[t] = logQ[(size_t)b * DD2 + t];
  __syncthreads();

  float q2p = 0.0f;
  for (int t = lane; t < DD2; t += 32) q2p += qb[t] * qb[t];
  const float q2 = wave_sum(q2p);

  for (int i = 0; i < M2; ++i) {
    const float* kv = logK + ((size_t)b * M2 + i) * DD2;
    float cp = 0.0f, kp = 0.0f;
    for (int t = lane; t < DD2; t += 32) {
      float kt = kv[t];
      cp += kt * qb[t];
      kp += kt * kt;
    }
    cp = wave_sum(cp);
    kp = wave_sum(kp);
    if (lane == 0) {
      float e = fmaxf(q2 + kp - 2.0f * cp, 0.0f);
      scores[i] = 1.0f / (1.0f + log1pf(e));
    }
  }
  __syncthreads();

  // softmax over i (axis -2 of reference, restricted to j=127)
  float mx = -3.4e38f;
  for (int i = lane; i < M2; i += 32) mx = fmaxf(mx, scores[i]);
  mx = wave_max(mx);
  float ssum = 0.0f;
  for (int i = lane; i < M2; i += 32) {
    float e = __expf(scores[i] - mx);
    scores[i] = e;
    ssum += e;
  }
  ssum = wave_sum(ssum);
  const float invs = 1.0f / ssum;
  __syncthreads();

  // mixed = sum_i prob_i * logV[b,i], pre-scaled by 2^-EXP_SQ for expm
  const float presc = invs * (1.0f / (float)(1 << EXP_SQ));
  for (int t = lane; t < DD2; t += 32) {
    float acc = 0.0f;
    for (int i = 0; i < M2; ++i)
      acc += scores[i] * logV[((size_t)b * M2 + i) * DD2 + t];
    Xb[t] = acc * presc;
  }
  __syncthreads();

  // exp(Xb) via Taylor (Horner) then EXP_SQ squarings — all 48^3 WMMA matmuls
  float cn = 1.0f / 5040.0f; // 1/(EXP_DEG-1)! with EXP_DEG=8
  for (int t = lane; t < DD2; t += 32)
    G[t] = (t % (DD + 1) == 0) ? (1.0f / 40320.0f) : 0.0f; // 1/8!
  __syncthreads();
  for (int n = EXP_DEG - 1; n >= 0; --n) {
    mm48(GN, Xb, G, lane);
    for (int t = lane; t < DD2; t += 32)
      GN[t] += (t % (DD + 1) == 0) ? cn : 0.0f;
    __syncthreads();
    float* tp = G; G = GN; GN = tp;
    cn *= (float)n; // c_{n-1} = c_n * n
  }
  for (int s = 0; s < EXP_SQ; ++s) {
    mm48(GN, G, G, lane);
    float* tp = G; G = GN; GN = tp;
  }

  for (int t = lane; t < DD2; t += 32) out[(size_t)b * DD2 + t] = G[t];
}

extern "C" void kernel_launch(void* const* d_in, const int* in_sizes, int n_in,
                              void* d_out, int out_size, void* d_ws, size_t ws_size,
                              hipStream_t stream) {
  (void)in_sizes; (void)n_in; (void)out_size; (void)ws_size;
  const float* x1 = (const float*)d_in[0];
  const float* x2 = (const float*)d_in[1];
  // d_in[2] = wq1 (unused: only j=127 of Q is needed, which comes from wq2)
  const float* wk1 = (const float*)d_in[3];
  const float* wv1 = (const float*)d_in[4];
  const float* wq2 = (const float*)d_in[5];
  const float* wk2 = (const float*)d_in[6];
  const float* wv2 = (const float*)d_in[7];

  float* logK = (float*)d_ws;                    // 4096*2304 floats
  float* logV = logK + (size_t)4096 * DD2;       // 4096*2304 floats
  float* logQ = logV + (size_t)4096 * DD2;       // 32*2304 floats

  bimap_logm_kernel<<<8224, 32, 0, stream>>>(x1, x2, wk1, wk2, wv1, wv2, wq2,
                                             logK, logV, logQ);
  attn_expm_kernel<<<32, 32, 0, stream>>>(logK, logV, logQ, (float*)d_out);
}